// POIEncoder_79018808312041
// MI455X (gfx1250) — compile-verified
//
#include <hip/hip_runtime.h>
#include <hip/hip_bf16.h>

// ---------------------------------------------------------------------------
// GCNConv (gcn_norm w/ self-loops) + bias + PReLU for MI455X (gfx1250).
//   deg  = 1 + sum_e w[e] at col[e];  dinv = rsqrt(deg)
//   h    = x @ W                      (fp32 WMMA 16x16x4)
//   out  = bias + dinv^2 * h          (self-loop, fused in GEMM epilogue)
//   out += dinv[s]*w*dinv[t] * h[s]   (edge scatter, f32 atomics, L2-resident)
//   out  = PReLU(out)
// ---------------------------------------------------------------------------

#define IN_C  128
#define HID_C 64

typedef __attribute__((ext_vector_type(2))) float v2f;
typedef __attribute__((ext_vector_type(8))) float v8f;

// ---- degree kernels --------------------------------------------------------

__global__ void gcn_init_deg(float* __restrict__ deg, int n) {
    int i = blockIdx.x * blockDim.x + threadIdx.x;
    if (i < n) deg[i] = 1.0f;                 // self-loop weight
}

__global__ void gcn_accum_deg(const int* __restrict__ col,
                              const float* __restrict__ ew,
                              float* __restrict__ deg, int e) {
    int i = blockIdx.x * blockDim.x + threadIdx.x;
    if (i < e) atomicAdd(&deg[col[i]], ew[i]);
}

__global__ void gcn_rsqrt_deg(float* __restrict__ deg, int n) {
    int i = blockIdx.x * blockDim.x + threadIdx.x;
    if (i < n) deg[i] = rsqrtf(deg[i]);
}

// ---- h = x @ W  with V_WMMA_F32_16X16X4_F32 --------------------------------
// Block = 256 threads = 8 waves. Wave w: col tile (w&3), row tile (w>>2).
// Block covers 32 rows x 64 cols. Epilogue writes h and seeds
// out = dinv^2 * h + bias (self-loop contribution + bias).

__global__ void __launch_bounds__(256)
gcn_gemm_wmma(const float* __restrict__ x, const float* __restrict__ W,
              const float* __restrict__ bias, const float* __restrict__ dinv,
              float* __restrict__ h, float* __restrict__ out, int nrows) {
    const int wave  = threadIdx.x >> 5;
    const int lane  = threadIdx.x & 31;
    const int n0    = (wave & 3) << 4;                 // 0,16,32,48
    const int row0  = (blockIdx.x * 2 + (wave >> 2)) << 4;
    if (row0 >= nrows) return;                         // wave-uniform exit (EXEC stays full)

    const int lhalf = lane >> 4;                       // 0: K=0,1  1: K=2,3
    const int l15   = lane & 15;

    int arow = row0 + l15;                             // A row for this lane
    if (arow >= nrows) arow = nrows - 1;               // clamp (stores predicated below)

    const float* __restrict__ ap = x + (size_t)arow * IN_C + (lhalf << 1);
    const float* __restrict__ bp = W + (size_t)(lhalf << 1) * HID_C + n0 + l15;

    v8f acc = {};
#pragma unroll
    for (int k0 = 0; k0 < IN_C; k0 += 4) {
        v2f a, b;
        a.x = ap[k0];
        a.y = ap[k0 + 1];
        b.x = bp[(size_t)k0 * HID_C];
        b.y = bp[(size_t)(k0 + 1) * HID_C];
        // D = A(16x4) * B(4x16) + C : emits v_wmma_f32_16x16x4_f32
        acc = __builtin_amdgcn_wmma_f32_16x16x4_f32(
            /*neg_a=*/false, a, /*neg_b=*/false, b,
            /*c_mod=*/(short)0, acc, /*reuse_a=*/false, /*reuse_b=*/false);
    }

    // D layout: vgpr i, lanes 0-15 -> M=row0+i, N=n0+lane; lanes 16-31 -> M=row0+8+i
    const int mbase = row0 + (lhalf << 3);
    const int n     = n0 + l15;
    const float bv  = bias[n];
#pragma unroll
    for (int i = 0; i < 8; ++i) {
        const int m = mbase + i;
        if (m < nrows) {
            const float v  = acc[i];
            const float di = dinv[m];
            h[(size_t)m * HID_C + n]   = v;
            out[(size_t)m * HID_C + n] = di * di * v + bv;   // self-loop + bias
        }
    }
}

// ---- edge scatter: one wave32 per edge, float2 per lane --------------------
// h and out are both L2-resident (25.6 MB each vs 192 MB L2), so the gather
// and the 64M global_atomic_add_f32 stay on-chip.

__global__ void __launch_bounds__(256)
gcn_scatter(const int* __restrict__ row, const int* __restrict__ col,
            const float* __restrict__ ew, const float* __restrict__ dinv,
            const float* __restrict__ h, float* __restrict__ out, int e) {
    const int wid  = blockIdx.x * 8 + (threadIdx.x >> 5);  // 8 waves / block
    if (wid >= e) return;
    const int lane = threadIdx.x & 31;

    const int   s    = row[wid];
    const int   t    = col[wid];
    const float coef = dinv[s] * ew[wid] * dinv[t];

    const v2f* __restrict__ hp = (const v2f*)(h + (size_t)s * HID_C);
    const v2f v = hp[lane];                                // coalesced 256B gather

    float* __restrict__ op = out + (size_t)t * HID_C + (lane << 1);
    atomicAdd(op,     coef * v.x);
    atomicAdd(op + 1, coef * v.y);
}

// ---- PReLU -----------------------------------------------------------------

__global__ void gcn_prelu(float* __restrict__ out, const float* __restrict__ a,
                          long long total) {
    long long i = (long long)blockIdx.x * blockDim.x + threadIdx.x;
    if (i < total) {
        const float v = out[i];
        out[i] = (v >= 0.0f) ? v : a[i & (HID_C - 1)] * v;
    }
}

// ---------------------------------------------------------------------------

extern "C" void kernel_launch(void* const* d_in, const int* in_sizes, int n_in,
                              void* d_out, int out_size, void* d_ws, size_t ws_size,
                              hipStream_t stream) {
    const float* x    = (const float*)d_in[0];
    const int*   ei   = (const int*)d_in[1];
    const float* ew   = (const float*)d_in[2];
    const float* W    = (const float*)d_in[3];
    const float* bias = (const float*)d_in[4];
    const float* pa   = (const float*)d_in[5];

    const int N = in_sizes[0] / IN_C;   // 100000
    const int E = in_sizes[2];          // 1000000
    const int* rowp = ei;               // edge_index[0] = sources
    const int* colp = ei + E;           // edge_index[1] = targets

    // Workspace: dinv[N] then h[N*HID] (~26 MB total)
    float* dinv = (float*)d_ws;
    float* h    = dinv + (((size_t)N + 3) & ~(size_t)3);
    float* out  = (float*)d_out;

    // 1) deg = 1 (self-loop), 2) deg += w at col, 3) dinv = rsqrt(deg)
    gcn_init_deg <<<(N + 255) / 256, 256, 0, stream>>>(dinv, N);
    gcn_accum_deg<<<(E + 255) / 256, 256, 0, stream>>>(colp, ew, dinv, E);
    gcn_rsqrt_deg<<<(N + 255) / 256, 256, 0, stream>>>(dinv, N);

    // 4) h = x @ W via WMMA; seed out = dinv^2*h + bias
    const int rowTiles = (N + 15) / 16;
    gcn_gemm_wmma<<<(rowTiles + 1) / 2, 256, 0, stream>>>(x, W, bias, dinv, h, out, N);

    // 5) edge scatter (one wave per edge)
    gcn_scatter<<<(E + 7) / 8, 256, 0, stream>>>(rowp, colp, ew, dinv, h, out, E);

    // 6) PReLU
    const long long total = (long long)N * HID_C;
    gcn_prelu<<<(unsigned)((total + 255) / 256), 256, 0, stream>>>(out, pa, total);
}